// GCN_LSTM_BI_37890201486067
// MI455X (gfx1250) — compile-verified
//
#include <hip/hip_runtime.h>
#include <hip/hip_bf16.h>
#include <math.h>

// ---------------------------------------------------------------------------
// GCN(3) + BiLSTM(2) + Linear for MI455X (gfx1250, wave32, WMMA).
// Dense GEMMs use V_WMMA_F32_16X16X4_F32; each wave owns a 16x64 output strip
// (4 accumulators) so every A-fragment load feeds 4 WMMAs.
// ---------------------------------------------------------------------------

#define NN     20000
#define EE     640000
#define IN_C   32
#define HID    256
#define G4H    1024      // 4*HID
#define D2H    512       // 2*HID
#define PRED   12
#define BN_EPS 1e-5f

typedef __attribute__((ext_vector_type(2))) float v2f;
typedef __attribute__((ext_vector_type(8))) float v8f;

// ---------------------------------------------------------------------------
// C[M x Nout] = A[M x K] @ W[Nout x K]^T (+ bias[Nout])
// One wave per 16x64 output strip (4 column tiles), K consumed 4/iter.
// VGPR layouts per CDNA5 ISA 7.12.2 (lane = 16*half + l):
//   A frag (v2f): a[j] = A[m0 + l][k + 2*half + j]          (contiguous f2)
//   B frag (v2f): b[j] = B[k + 2*half + j][n0 + l]
//               = W[n0 + l][k + 2*half + j]                  (contiguous f2)
//   C/D  (v8f):  c[r] = C[m0 + r + 8*half][n0 + l]
// Requires M%16==0, Nout%64==0, K%4==0 (true for all call sites here).
// ---------------------------------------------------------------------------
__global__ void __launch_bounds__(256)
wmma_gemm_nt(const float* __restrict__ A, const float* __restrict__ W,
             const float* __restrict__ bias, float* __restrict__ C,
             int M, int K, int Nout)
{
    const int lane  = threadIdx.x & 31;
    const int wave  = threadIdx.x >> 5;
    const int half  = lane >> 4;        // 0: lanes 0-15, 1: lanes 16-31
    const int l     = lane & 15;
    const int stripsN = Nout >> 6;      // 64-wide strips
    const int totalStrips = (M >> 4) * stripsN;

    int strip = blockIdx.x * (blockDim.x >> 5) + wave;
    if (strip >= totalStrips) return;   // wave-uniform: EXEC all-1s for WMMA

    const int sm = strip / stripsN;
    const int sn = strip - sm * stripsN;
    const int m0 = sm << 4;
    const int n0 = sn << 6;

    const float* aRow  = A + (size_t)(m0 + l) * K + 2 * half;
    const float* wRow0 = W + (size_t)(n0 +  0 + l) * K + 2 * half;
    const float* wRow1 = W + (size_t)(n0 + 16 + l) * K + 2 * half;
    const float* wRow2 = W + (size_t)(n0 + 32 + l) * K + 2 * half;
    const float* wRow3 = W + (size_t)(n0 + 48 + l) * K + 2 * half;

    v8f c0 = {}, c1 = {}, c2 = {}, c3 = {};
    #pragma unroll 2
    for (int k = 0; k < K; k += 4) {
        v2f a  = *(const v2f*)(aRow  + k);
        v2f b0 = *(const v2f*)(wRow0 + k);
        v2f b1 = *(const v2f*)(wRow1 + k);
        v2f b2 = *(const v2f*)(wRow2 + k);
        v2f b3 = *(const v2f*)(wRow3 + k);
        // (neg_a, A, neg_b, B, c_mod, C, reuse_a, reuse_b)
        c0 = __builtin_amdgcn_wmma_f32_16x16x4_f32(false, a, false, b0, (short)0, c0, false, false);
        c1 = __builtin_amdgcn_wmma_f32_16x16x4_f32(false, a, false, b1, (short)0, c1, false, false);
        c2 = __builtin_amdgcn_wmma_f32_16x16x4_f32(false, a, false, b2, (short)0, c2, false, false);
        c3 = __builtin_amdgcn_wmma_f32_16x16x4_f32(false, a, false, b3, (short)0, c3, false, false);
    }

    float* cBase = C + (size_t)(m0 + 8 * half) * Nout + n0 + l;
    const float bv0 = bias ? bias[n0 +  0 + l] : 0.0f;
    const float bv1 = bias ? bias[n0 + 16 + l] : 0.0f;
    const float bv2 = bias ? bias[n0 + 32 + l] : 0.0f;
    const float bv3 = bias ? bias[n0 + 48 + l] : 0.0f;
    #pragma unroll
    for (int r = 0; r < 8; ++r) {
        float* cr = cBase + (size_t)r * Nout;
        cr[ 0] = c0[r] + bv0;
        cr[16] = c1[r] + bv1;
        cr[32] = c2[r] + bv2;
        cr[48] = c3[r] + bv3;
    }
}

// ---------------------------------------------------------------------------
__global__ void fill_f32(float* __restrict__ p, size_t n, float v)
{
    size_t i = (size_t)blockIdx.x * blockDim.x + threadIdx.x;
    if (i < n) p[i] = v;
}

// msg = h[src] * w ; agg[dst] += msg   (thread = one edge x 4 channels)
__global__ void scatter_edges(const float* __restrict__ h,
                              const long long* __restrict__ src,
                              const long long* __restrict__ dst,
                              const float* __restrict__ ew,
                              float* __restrict__ agg)
{
    int t = blockIdx.x * blockDim.x + threadIdx.x;
    const int per = HID >> 2;                 // 64 float4 groups per edge
    int e  = t / per;
    int c4 = (t - e * per) << 2;
    if (e >= EE) return;
    const long long s = src[e];
    const long long d = dst[e];
    const float w = ew[e];
    const float4 hv = *(const float4*)(h + (size_t)s * HID + c4);
    float* ap = agg + (size_t)d * HID + c4;
    atomicAdd(ap + 0, hv.x * w);
    atomicAdd(ap + 1, hv.y * w);
    atomicAdd(ap + 2, hv.z * w);
    atomicAdd(ap + 3, hv.w * w);
}

// per-channel partial sums for batch-norm (block = 128 rows, thread = channel)
__global__ void __launch_bounds__(HID)
bn_partial(const float* __restrict__ x, float* __restrict__ sum,
           float* __restrict__ sq, int rowsPerBlock)
{
    const int c  = threadIdx.x;
    const int r0 = blockIdx.x * rowsPerBlock;
    const int r1 = min(r0 + rowsPerBlock, NN);
    float s = 0.f, ss = 0.f;
    for (int r = r0; r < r1; ++r) {
        float v = x[(size_t)r * HID + c];
        s  += v;
        ss += v * v;
    }
    atomicAdd(&sum[c], s);
    atomicAdd(&sq[c],  ss);
}

// fold mean/var + gamma/beta into scale/shift (GCN bias cancels through BN)
__global__ void __launch_bounds__(HID)
bn_finalize(const float* __restrict__ sum, const float* __restrict__ sq,
            const float* __restrict__ g, const float* __restrict__ b,
            float* __restrict__ scale, float* __restrict__ shift)
{
    const int c = threadIdx.x;
    const float invN = 1.0f / (float)NN;
    const float mean = sum[c] * invN;
    const float var  = sq[c] * invN - mean * mean;
    const float sc   = g[c] * rsqrtf(var + BN_EPS);
    scale[c] = sc;
    shift[c] = b[c] - mean * sc;
}

// x_out = relu(agg * scale + shift)
__global__ void bn_apply_relu(const float* __restrict__ agg,
                              const float* __restrict__ scale,
                              const float* __restrict__ shift,
                              float* __restrict__ xo)
{
    int t = blockIdx.x * blockDim.x + threadIdx.x;
    if (t >= NN * HID) return;
    int c = t & (HID - 1);
    float v = fmaf(agg[t], scale[c], shift[c]);
    xo[t] = fmaxf(v, 0.0f);
}

// ---------------------------------------------------------------------------
// Bidirectional LSTM scan over the 20000-step sequence.
// grid = 2 blocks (dir 0 = fwd, dir 1 = bwd), 1024 threads (thread n = gate n).
// pre[t] already holds seq[t] @ Wih^T + b. gates: [i | f | g | o].
// The 256-deep dot product uses 4 independent accumulators to cut the serial
// FMA chain (per-step latency dominates total runtime).
// ---------------------------------------------------------------------------
__global__ void __launch_bounds__(G4H)
lstm_scan_bidir(const float* __restrict__ preF, const float* __restrict__ preB,
                const float* __restrict__ WhhF, const float* __restrict__ WhhB,
                float* __restrict__ outSeq)
{
    __shared__ float h_sh[HID];
    __shared__ float g_sh[G4H];

    const int n   = threadIdx.x;
    const int dir = blockIdx.x;
    const float* __restrict__ pre = dir ? preB : preF;
    const float* __restrict__ Whh = dir ? WhhB : WhhF;
    const float* __restrict__ wrow = Whh + (size_t)n * HID;

    float cst = 0.0f;                 // cell state (threads 0..255 own it)
    if (n < HID) h_sh[n] = 0.0f;
    __syncthreads();

    for (int it = 0; it < NN; ++it) {
        const int t = dir ? (NN - 1 - it) : it;
        // prefetch next step's pre row (global_prefetch_b8)
        if (it + 1 < NN) {
            const int tn = dir ? (t - 1) : (t + 1);
            __builtin_prefetch(pre + (size_t)tn * G4H + n, 0, 1);
        }
        float a0 = 0.f, a1 = 0.f, a2 = 0.f, a3 = 0.f;
        #pragma unroll 4
        for (int k = 0; k < HID; k += 16) {
            float4 wa = *(const float4*)(wrow + k);
            float4 wb = *(const float4*)(wrow + k + 4);
            float4 wc = *(const float4*)(wrow + k + 8);
            float4 wd = *(const float4*)(wrow + k + 12);
            a0 = fmaf(h_sh[k +  0], wa.x, a0); a0 = fmaf(h_sh[k +  1], wa.y, a0);
            a0 = fmaf(h_sh[k +  2], wa.z, a0); a0 = fmaf(h_sh[k +  3], wa.w, a0);
            a1 = fmaf(h_sh[k +  4], wb.x, a1); a1 = fmaf(h_sh[k +  5], wb.y, a1);
            a1 = fmaf(h_sh[k +  6], wb.z, a1); a1 = fmaf(h_sh[k +  7], wb.w, a1);
            a2 = fmaf(h_sh[k +  8], wc.x, a2); a2 = fmaf(h_sh[k +  9], wc.y, a2);
            a2 = fmaf(h_sh[k + 10], wc.z, a2); a2 = fmaf(h_sh[k + 11], wc.w, a2);
            a3 = fmaf(h_sh[k + 12], wd.x, a3); a3 = fmaf(h_sh[k + 13], wd.y, a3);
            a3 = fmaf(h_sh[k + 14], wd.z, a3); a3 = fmaf(h_sh[k + 15], wd.w, a3);
        }
        float acc = pre[(size_t)t * G4H + n] + ((a0 + a1) + (a2 + a3));
        float gv;
        if (n >= 2 * HID && n < 3 * HID) gv = tanhf(acc);            // g gate
        else                             gv = 1.0f / (1.0f + __expf(-acc));
        g_sh[n] = gv;
        __syncthreads();              // gates visible; h_sh reads complete
        if (n < HID) {
            cst = fmaf(g_sh[HID + n], cst, g_sh[n] * g_sh[2 * HID + n]);
            float hv = g_sh[3 * HID + n] * tanhf(cst);
            h_sh[n] = hv;
            outSeq[(size_t)t * D2H + dir * HID + n] = hv;
        }
        __syncthreads();              // new h visible for next step
    }
}

// out[m][p] = seq[m] . out_W[p] + out_b[p]   (p < 12, 16 threads per row)
__global__ void out_proj(const float* __restrict__ seq,
                         const float* __restrict__ W,
                         const float* __restrict__ b,
                         float* __restrict__ out)
{
    int t = blockIdx.x * blockDim.x + threadIdx.x;
    int m = t >> 4;
    int p = t & 15;
    if (m >= NN || p >= PRED) return;
    const float* srow = seq + (size_t)m * D2H;
    const float* wrow = W + (size_t)p * D2H;
    float acc = b[p];
    #pragma unroll 8
    for (int k = 0; k < D2H; k += 4) {
        acc = fmaf(srow[k + 0], wrow[k + 0], acc);
        acc = fmaf(srow[k + 1], wrow[k + 1], acc);
        acc = fmaf(srow[k + 2], wrow[k + 2], acc);
        acc = fmaf(srow[k + 3], wrow[k + 3], acc);
    }
    out[(size_t)m * PRED + p] = acc;
}

// ---------------------------------------------------------------------------
static inline int cdiv(long long a, long long b) { return (int)((a + b - 1) / b); }

extern "C" void kernel_launch(void* const* d_in, const int* in_sizes, int n_in,
                              void* d_out, int out_size, void* d_ws, size_t ws_size,
                              hipStream_t stream)
{
    // ---- inputs (setup_inputs dict order, lists flattened) ----
    const float*      x0   = (const float*)d_in[0];           // N x 32
    const long long*  ei   = (const long long*)d_in[1];       // 2 x E (int64)
    const float*      ew   = (const float*)d_in[2];           // E
    const float*      gW[3]  = {(const float*)d_in[3],  (const float*)d_in[4],  (const float*)d_in[5]};
    // gcn_bs (6..8) cancel exactly through batch-norm; skipped.
    const float*      bng[3] = {(const float*)d_in[9],  (const float*)d_in[10], (const float*)d_in[11]};
    const float*      bnb[3] = {(const float*)d_in[12], (const float*)d_in[13], (const float*)d_in[14]};
    const float*      Wih[4] = {(const float*)d_in[15], (const float*)d_in[16], (const float*)d_in[17], (const float*)d_in[18]};
    const float*      Whh[4] = {(const float*)d_in[19], (const float*)d_in[20], (const float*)d_in[21], (const float*)d_in[22]};
    const float*      lb[4]  = {(const float*)d_in[23], (const float*)d_in[24], (const float*)d_in[25], (const float*)d_in[26]};
    const float*      outW = (const float*)d_in[27];          // 12 x 512
    const float*      outb = (const float*)d_in[28];          // 12
    float*            out  = (float*)d_out;                   // N x 12

    const long long* src = ei;        // edge_index[0]
    const long long* dst = ei + EE;   // edge_index[1]

    // ---- workspace layout (floats) ----
    float* ws   = (float*)d_ws;
    float* bufA = ws;                         // N*512  seq ping
    float* bufB = bufA + (size_t)NN * D2H;    // N*512  seq pong
    float* preF = bufB + (size_t)NN * D2H;    // N*1024 (also GCN h scratch)
    float* preB = preF + (size_t)NN * G4H;    // N*1024 (also GCN agg scratch)
    float* st   = preB + (size_t)NN * G4H;    // 1024: sum | sq | scale | shift
    (void)ws_size; (void)in_sizes; (void)n_in; (void)out_size;

    float* h   = preF;   // GCN GEMM output (N*256 region)
    float* agg = preB;   // GCN aggregation (N*256 region)

    const int TPB = 256;
    const int ROWS_PER_BN_BLOCK = 128;

    // ---- GCN stack ----
    const float* xin = x0;
    int K = IN_C;
    float* gcnOut[3] = {bufA, bufB, bufA};    // layer outputs (N*256 each)
    for (int i = 0; i < 3; ++i) {
        int strips = (NN / 16) * (HID / 64);
        wmma_gemm_nt<<<cdiv(strips, TPB / 32), TPB, 0, stream>>>(
            xin, gW[i], nullptr, h, NN, K, HID);

        fill_f32<<<cdiv((long long)NN * HID, TPB), TPB, 0, stream>>>(
            agg, (size_t)NN * HID, 0.0f);
        scatter_edges<<<cdiv((long long)EE * (HID / 4), TPB), TPB, 0, stream>>>(
            h, src, dst, ew, agg);

        fill_f32<<<cdiv(2 * HID, TPB), TPB, 0, stream>>>(st, 2 * HID, 0.0f);
        bn_partial<<<cdiv(NN, ROWS_PER_BN_BLOCK), HID, 0, stream>>>(
            agg, st, st + HID, ROWS_PER_BN_BLOCK);
        bn_finalize<<<1, HID, 0, stream>>>(
            st, st + HID, bng[i], bnb[i], st + 2 * HID, st + 3 * HID);
        bn_apply_relu<<<cdiv((long long)NN * HID, TPB), TPB, 0, stream>>>(
            agg, st + 2 * HID, st + 3 * HID, gcnOut[i]);

        xin = gcnOut[i];
        K = HID;
    }
    // seq0 = bufA (N x 256)

    // ---- BiLSTM layer 0 (D=256): bufA -> bufB ----
    {
        int strips = (NN / 16) * (G4H / 64);
        wmma_gemm_nt<<<cdiv(strips, TPB / 32), TPB, 0, stream>>>(
            bufA, Wih[0], lb[0], preF, NN, HID, G4H);
        wmma_gemm_nt<<<cdiv(strips, TPB / 32), TPB, 0, stream>>>(
            bufA, Wih[1], lb[1], preB, NN, HID, G4H);
        lstm_scan_bidir<<<2, G4H, 0, stream>>>(preF, preB, Whh[0], Whh[1], bufB);
    }
    // ---- BiLSTM layer 1 (D=512): bufB -> bufA ----
    {
        int strips = (NN / 16) * (G4H / 64);
        wmma_gemm_nt<<<cdiv(strips, TPB / 32), TPB, 0, stream>>>(
            bufB, Wih[2], lb[2], preF, NN, D2H, G4H);
        wmma_gemm_nt<<<cdiv(strips, TPB / 32), TPB, 0, stream>>>(
            bufB, Wih[3], lb[3], preB, NN, D2H, G4H);
        lstm_scan_bidir<<<2, G4H, 0, stream>>>(preF, preB, Whh[2], Whh[3], bufA);
    }

    // ---- output projection ----
    out_proj<<<cdiv((long long)NN * 16, TPB), TPB, 0, stream>>>(bufA, outW, outb, out);
}